// EfficientGraphConv_4398046511507
// MI455X (gfx1250) — compile-verified
//
#include <hip/hip_runtime.h>
#include <math.h>
#include <float.h>

#define N_NODES 50000
#define N_EDGES 800000
#define IN_C    128
#define OUT_C   128
#define HEADS   8
#define BASES   4
#define HEAD_D  16
#define BD      (BASES * HEAD_D)   // 64

typedef float v2f __attribute__((ext_vector_type(2)));
typedef float v8f __attribute__((ext_vector_type(8)));

// ---------------------------------------------------------------------------
// sign-aware exact float atomic max (int order for +, uint order for -)
// ---------------------------------------------------------------------------
__device__ __forceinline__ void atomicMaxFloat(float* addr, float val) {
  unsigned bits = __float_as_uint(val);
  if ((bits & 0x80000000u) == 0u) {
    atomicMax((int*)addr, (int)bits);
  } else {
    atomicMin((unsigned int*)addr, bits);
  }
}

// ---------------------------------------------------------------------------
// K0: init aggregator buffer [N][4][64] (part 3 = max -> -inf) and cnt
// ---------------------------------------------------------------------------
__global__ __launch_bounds__(256) void k_init(float* __restrict__ agg,
                                              unsigned* __restrict__ cnt) {
  int i = blockIdx.x * 256 + threadIdx.x;
  if (i < N_NODES * 256) {
    int part = (i >> 6) & 3;
    agg[i] = (part == 3) ? -INFINITY : 0.0f;
  }
  if (i < N_NODES) cnt[i] = 0u;
}

// ---------------------------------------------------------------------------
// K1: in-degree of col
// ---------------------------------------------------------------------------
__global__ __launch_bounds__(256) void k_degree(const int* __restrict__ col,
                                                unsigned* __restrict__ cnt) {
  int e = blockIdx.x * 256 + threadIdx.x;
  if (e < N_EDGES) atomicAdd(&cnt[col[e]], 1u);
}

// ---------------------------------------------------------------------------
// K2: dinv = rsqrt(deg + self-loop)
// ---------------------------------------------------------------------------
__global__ __launch_bounds__(256) void k_dinv(const unsigned* __restrict__ cnt,
                                              float* __restrict__ dinv) {
  int n = blockIdx.x * 256 + threadIdx.x;
  if (n < N_NODES) dinv[n] = rsqrtf((float)cnt[n] + 1.0f);
}

// ---------------------------------------------------------------------------
// K3: fused GEMM (bases = x @ Wb, logits = x @ Wc + b -> per-head softmax)
//     using V_WMMA_F32_16X16X4_F32 (exact fp32 matrix path on CDNA5)
//     block = 128 threads = 4 waves, 64 rows of x staged in LDS.
// ---------------------------------------------------------------------------
__global__ __launch_bounds__(128) void k_gemm(const float* __restrict__ x,
                                              const float* __restrict__ bW,
                                              const float* __restrict__ cW,
                                              const float* __restrict__ cb,
                                              float* __restrict__ bases,
                                              float* __restrict__ wbuf) {
  __shared__ float xs[64][132];  // pad 132: stride%64 = 4 -> conflict-free A reads

  const int tid  = threadIdx.x;
  const int row0 = blockIdx.x * 64;

  // cooperative coalesced load of 64x128 fp32 tile (row-clamped for tail block)
  for (int t = 0; t < 16; ++t) {
    int idx = t * 128 + tid;          // 0..2047 float4 slots
    int r   = idx >> 5;               // 0..63
    int c4  = (idx & 31) * 4;         // 0..124
    int gr  = row0 + r;
    if (gr >= N_NODES) gr = N_NODES - 1;
    const float4 v = *(const float4*)(x + (size_t)gr * IN_C + c4);
    xs[r][c4 + 0] = v.x; xs[r][c4 + 1] = v.y;
    xs[r][c4 + 2] = v.z; xs[r][c4 + 3] = v.w;
  }
  __syncthreads();

  const int wave = tid >> 5;
  const int lane = tid & 31;
  const int nlo  = lane & 15;          // col within 16-wide tile / M within A
  const int half = lane >> 4;          // 0: lanes 0-15, 1: lanes 16-31
  const int kb   = half * 2;           // A/B frag K offset per ISA layout
  const int lr   = wave * 16 + nlo;    // local row feeding the A fragment

  for (int nt = 0; nt < 12; ++nt) {
    v8f acc = {0.f, 0.f, 0.f, 0.f, 0.f, 0.f, 0.f, 0.f};

    if (nt < 4) {
      // bases_W[nt] : [IN_C][HEAD_D]
      const float* W = bW + nt * (IN_C * HEAD_D);
#pragma unroll
      for (int ks = 0; ks < 32; ++ks) {
        const int k = ks * 4 + kb;
        v2f a = *(const v2f*)&xs[lr][k];                    // ds_load_b64
        v2f b = { W[k * HEAD_D + nlo], W[(k + 1) * HEAD_D + nlo] };
        acc = __builtin_amdgcn_wmma_f32_16x16x4_f32(
            false, a, false, b, (short)0, acc, false, false);
      }
#pragma unroll
      for (int r = 0; r < 8; ++r) {
        const int m  = r + (half << 3);
        const int gr = row0 + wave * 16 + m;
        if (gr < N_NODES) bases[(size_t)gr * BD + nt * 16 + nlo] = acc[r];
      }
    } else {
      const int   head = nt - 4;               // one 16-wide softmax group
      const float* W   = cW + head * 16;       // comb_W [IN_C][128]
#pragma unroll
      for (int ks = 0; ks < 32; ++ks) {
        const int k = ks * 4 + kb;
        v2f a = *(const v2f*)&xs[lr][k];
        v2f b = { W[k * OUT_C + nlo], W[(k + 1) * OUT_C + nlo] };
        acc = __builtin_amdgcn_wmma_f32_16x16x4_f32(
            false, a, false, b, (short)0, acc, false, false);
      }
      const float bb = cb[head * 16 + nlo];
#pragma unroll
      for (int r = 0; r < 8; ++r) {
        float v = acc[r] + bb;
        // softmax over the 16 lanes of each half-wave (one matrix row each)
        float mx = v;
        mx = fmaxf(mx, __shfl_xor(mx, 1));
        mx = fmaxf(mx, __shfl_xor(mx, 2));
        mx = fmaxf(mx, __shfl_xor(mx, 4));
        mx = fmaxf(mx, __shfl_xor(mx, 8));
        float e = __expf(v - mx);
        float s = e;
        s += __shfl_xor(s, 1);
        s += __shfl_xor(s, 2);
        s += __shfl_xor(s, 4);
        s += __shfl_xor(s, 8);
        const float wv = e / s;
        const int m  = r + (half << 3);
        const int gr = row0 + wave * 16 + m;
        if (gr < N_NODES) wbuf[(size_t)gr * OUT_C + head * 16 + nlo] = wv;
      }
    }
  }
}

// ---------------------------------------------------------------------------
// K4: edge scatter. 16 lanes per edge, float4 of bases per lane,
//     16 global f32 atomics per lane into node-major agg[n][4][64].
// ---------------------------------------------------------------------------
__global__ __launch_bounds__(256) void k_scatter(const int* __restrict__ ei,
                                                 const float* __restrict__ bases,
                                                 const float* __restrict__ dinv,
                                                 float* __restrict__ agg) {
  const long long g = (long long)blockIdx.x * 256 + threadIdx.x;
  const int e = (int)(g >> 4);
  if (e >= N_EDGES) return;
  const int q = (int)(g & 15);

  const int row = ei[e];
  const int col = ei[N_EDGES + e];
  if (q == 0 && e + 4096 < N_EDGES) {
    __builtin_prefetch(ei + e + 4096, 0, 0);               // global_prefetch_b8
    __builtin_prefetch(ei + N_EDGES + e + 4096, 0, 0);
  }

  const float  dr = dinv[row];
  const float4 v  = *(const float4*)(bases + (size_t)row * BD + q * 4);
  float* base     = agg + (size_t)col * 256;

  const float vals[4] = {v.x, v.y, v.z, v.w};
#pragma unroll
  for (int c = 0; c < 4; ++c) {
    const int   idx = q * 4 + c;
    const float t   = vals[c];
    atomicAdd(base + idx,       dr * t);   // symadd numerator (dinv[row]*m)
    atomicAdd(base + 64 + idx,  t);        // sum
    atomicAdd(base + 128 + idx, t * t);    // sum of squares
    atomicMaxFloat(base + 192 + idx, t);   // max
  }
}

// ---------------------------------------------------------------------------
// K5: per-node combine: z[n,h,d] = sum_{b,a} w[n,h,b,a] * agg_a[n,b,d] + bias
//     one node per 128-thread block, staged in LDS.
// ---------------------------------------------------------------------------
__global__ __launch_bounds__(128) void k_combine(const float* __restrict__ agg,
                                                 const float* __restrict__ bases,
                                                 const float* __restrict__ wbuf,
                                                 const unsigned* __restrict__ cnt,
                                                 const float* __restrict__ dinv,
                                                 const float* __restrict__ bias,
                                                 float* __restrict__ out) {
  __shared__ float ws_[128];
  __shared__ float as_[256];
  __shared__ float bs_[64];

  const int n   = blockIdx.x;
  const int tid = threadIdx.x;

  ws_[tid]       = wbuf[(size_t)n * 128 + tid];
  as_[tid]       = agg[(size_t)n * 256 + tid];
  as_[tid + 128] = agg[(size_t)n * 256 + 128 + tid];
  if (tid < 64) bs_[tid] = bases[(size_t)n * 64 + tid];
  __syncthreads();

  const float cf = (float)cnt[n];
  const float ic = 1.0f / fmaxf(cf, 1.0f);
  const float dv = dinv[n];
  const int h = tid >> 4;
  const int d = tid & 15;

  float o = bias[tid];
#pragma unroll
  for (int b = 0; b < 4; ++b) {
    const int   idx  = b * 16 + d;
    const float sym  = dv * (as_[idx] + dv * bs_[idx]);      // + self-loop term
    const float s    = as_[64 + idx];
    const float mean = s * ic;
    const float var  = as_[128 + idx] * ic - mean * mean;
    const float sd   = sqrtf(fmaxf(var, 0.0f) + 1e-5f);
    const float mx   = (cf > 0.0f) ? as_[192 + idx] : 0.0f;  // empty segs -> 0
    const float* wp  = ws_ + h * 16 + b * 4;
    o += wp[0] * sym + wp[1] * mean + wp[2] * mx + wp[3] * sd;
  }
  out[(size_t)n * 128 + tid] = o;
}

// ---------------------------------------------------------------------------
extern "C" void kernel_launch(void* const* d_in, const int* in_sizes, int n_in,
                              void* d_out, int out_size, void* d_ws, size_t ws_size,
                              hipStream_t stream) {
  (void)in_sizes; (void)n_in; (void)out_size; (void)ws_size;
  const float* x    = (const float*)d_in[0];
  const int*   ei   = (const int*)d_in[1];   // [2, E]
  const float* bW   = (const float*)d_in[2]; // [BASES, IN_C, HEAD_D]
  const float* cW   = (const float*)d_in[3]; // [IN_C, 128]
  const float* cb   = (const float*)d_in[4]; // [128]
  const float* bias = (const float*)d_in[5]; // [128]
  float*       out  = (float*)d_out;

  char* ws = (char*)d_ws;
  float*    agg   = (float*)ws;  ws += (size_t)N_NODES * 256 * sizeof(float);
  float*    bases = (float*)ws;  ws += (size_t)N_NODES * BD * sizeof(float);
  float*    wbuf  = (float*)ws;  ws += (size_t)N_NODES * OUT_C * sizeof(float);
  unsigned* cnt   = (unsigned*)ws; ws += (size_t)N_NODES * sizeof(unsigned);
  float*    dinv  = (float*)ws;

  k_init<<<N_NODES, 256, 0, stream>>>(agg, cnt);
  k_degree<<<(N_EDGES + 255) / 256, 256, 0, stream>>>(ei + N_EDGES, cnt);
  k_gemm<<<(N_NODES + 63) / 64, 128, 0, stream>>>(x, bW, cW, cb, bases, wbuf);
  k_dinv<<<(N_NODES + 255) / 256, 256, 0, stream>>>(cnt, dinv);
  {
    const long long work = (long long)N_EDGES * 16;
    k_scatter<<<(unsigned)((work + 255) / 256), 256, 0, stream>>>(ei, bases, dinv, agg);
  }
  k_combine<<<N_NODES, 128, 0, stream>>>(agg, bases, wbuf, cnt, dinv, bias, out);
}